// CrossAttention_28363964022992
// MI455X (gfx1250) — compile-verified
//
#include <hip/hip_runtime.h>
#include <hip/hip_bf16.h>

// CrossAttention fused:  out[b] = (scale * (fpsf[b] @ Wq^T reshaped) @ Wk_h) @ fimg[b]
//   M[b,h,c] = scale * sum_d q[b,h,d] * Wk[h*64+d, c]      (tiny, VALU kernel -> d_ws)
//   out[b]   = M[b] (64x64)  @  fimg[b] (64x16384)         (WMMA f32 16x16x4 kernel)

#define B_      4
#define C_      64       // == HEADS
#define HW_     16384    // 128*128
#define DHEAD_  64

typedef __attribute__((ext_vector_type(2))) float v2f;
typedef __attribute__((ext_vector_type(8))) float v8f;

// ---------------------------------------------------------------------------
// Kernel 1: M[b,h,c] = 0.125 * sum_d ( sum_c' fpsf[b,c'] Wq[h*64+d,c'] ) * Wk[h*64+d,c]
// grid (HEADS, B), block 64 (2 waves). Negligible cost (~4 MFLOP total).
// ---------------------------------------------------------------------------
__global__ __launch_bounds__(64) void ca_make_m(const float* __restrict__ fpsf,
                                                const float* __restrict__ Wq,
                                                const float* __restrict__ Wk,
                                                float* __restrict__ M) {
  const int h = blockIdx.x;
  const int b = blockIdx.y;
  const int t = threadIdx.x;          // 0..63

  __shared__ float qs[DHEAD_];

  // q[b,h,d] for d = t
  {
    const float* wq = Wq + (size_t)(h * DHEAD_ + t) * C_;
    const float* fp = fpsf + (size_t)b * C_;
    float s = 0.f;
#pragma unroll
    for (int c = 0; c < C_; ++c) s = fmaf(fp[c], wq[c], s);
    qs[t] = s;
  }
  __syncthreads();

  // M[b,h,c] for c = t  (lane index == c -> coalesced Wk reads)
  float m = 0.f;
#pragma unroll 8
  for (int d = 0; d < DHEAD_; ++d)
    m = fmaf(qs[d], Wk[(size_t)(h * DHEAD_ + d) * C_ + t], m);

  M[((size_t)b * C_ + h) * C_ + t] = m * 0.125f;  // scale = 64^-0.5
}

// ---------------------------------------------------------------------------
// Kernel 2: out[b] = M[b] @ fimg[b]   (64x64 @ 64x16384), fp32 WMMA 16x16x4.
// grid (HW/128, B), block 256 = 8 waves. Wave w: columns j0 = bx*128 + w*16,
// all 4 row-tiles (64 output rows). fimg is streamed exactly once.
// M[b] staged in LDS with row stride 68 floats -> conflict-free ds_load_b64:
//   bank(lane) = (68*col16 + k0 + 2*half) & 63 = (4*col16 + koff) & 63,
//   32 lanes x 2 banks (b64) cover 64 distinct banks.
// ---------------------------------------------------------------------------
#define MPAD 68

__global__ __launch_bounds__(256) void ca_gemm(const float* __restrict__ M,
                                               const float* __restrict__ img,
                                               float* __restrict__ out) {
  __shared__ float Ml[C_ * MPAD];

  const int b = blockIdx.y;

  // Cooperative stage of M[b] (4096 floats) into padded LDS.
  {
    const float* Mb = M + (size_t)b * C_ * C_;
#pragma unroll
    for (int i = 0; i < 16; ++i) {
      int idx = threadIdx.x + 256 * i;          // 0..4095
      int r = idx >> 6, c = idx & 63;
      Ml[r * MPAD + c] = Mb[idx];
    }
  }
  __syncthreads();

  const int lane  = threadIdx.x & 31;
  const int wave  = threadIdx.x >> 5;
  const int half  = lane >> 4;                  // 0: K lo pair, 1: K hi pair
  const int col16 = lane & 15;                  // A: M-row within tile; B/D: N col
  const int j     = blockIdx.x * 128 + wave * 16 + col16;

  const float* imgb = img + (size_t)b * C_ * HW_ + j;   // column j, row stride HW_

  v8f acc0 = {}, acc1 = {}, acc2 = {}, acc3 = {};

#pragma unroll
  for (int k0 = 0; k0 < C_; k0 += 4) {
    const int koff = k0 + 2 * half;

    // B fragment (4x16 of fimg): lane holds rows {koff, koff+1} at column j.
    v2f bv;
    bv.x = imgb[(size_t)koff * HW_];
    bv.y = imgb[(size_t)(koff + 1) * HW_];

    // A fragments (16x4 of M) for the 4 row-tiles, from LDS (ds_load_b64).
    const float* lp = &Ml[col16 * MPAD + koff];
    v2f a0 = *(const v2f*)(lp + 0 * 16 * MPAD);
    v2f a1 = *(const v2f*)(lp + 1 * 16 * MPAD);
    v2f a2 = *(const v2f*)(lp + 2 * 16 * MPAD);
    v2f a3 = *(const v2f*)(lp + 3 * 16 * MPAD);

    acc0 = __builtin_amdgcn_wmma_f32_16x16x4_f32(false, a0, false, bv, (short)0, acc0, false, false);
    acc1 = __builtin_amdgcn_wmma_f32_16x16x4_f32(false, a1, false, bv, (short)0, acc1, false, false);
    acc2 = __builtin_amdgcn_wmma_f32_16x16x4_f32(false, a2, false, bv, (short)0, acc2, false, false);
    acc3 = __builtin_amdgcn_wmma_f32_16x16x4_f32(false, a3, false, bv, (short)0, acc3, false, false);
  }

  // D layout: VGPR r, lanes 0-15 -> row r, lanes 16-31 -> row r+8 (within tile).
  float* ob = out + (size_t)b * C_ * HW_ + j;
#pragma unroll
  for (int r = 0; r < 8; ++r) {
    const size_t row = (size_t)(r + 8 * half);
    ob[(row +  0) * HW_] = acc0[r];
    ob[(row + 16) * HW_] = acc1[r];
    ob[(row + 32) * HW_] = acc2[r];
    ob[(row + 48) * HW_] = acc3[r];
  }
}

// ---------------------------------------------------------------------------
extern "C" void kernel_launch(void* const* d_in, const int* in_sizes, int n_in,
                              void* d_out, int out_size, void* d_ws, size_t ws_size,
                              hipStream_t stream) {
  const float* fpsf = (const float*)d_in[0];   // [4,64]
  const float* fimg = (const float*)d_in[1];   // [4,64,128,128]
  const float* Wq   = (const float*)d_in[2];   // [4096,64]
  const float* Wk   = (const float*)d_in[3];   // [4096,64]
  float* out = (float*)d_out;                  // [4,64,128,128]
  float* M   = (float*)d_ws;                   // 4*64*64 floats = 64 KB scratch

  ca_make_m<<<dim3(C_, B_), 64, 0, stream>>>(fpsf, Wq, Wk, M);
  ca_gemm<<<dim3(HW_ / 128, B_), 256, 0, stream>>>(M, fimg, out);
}